// ItemCodeDPQ_85023172591850
// MI455X (gfx1250) — compile-verified
//
#include <hip/hip_runtime.h>
#include <hip/hip_bf16.h>
#include <stdint.h>

// Product-quantized embedding gather for MI455X (gfx1250, wave32).
//
// Per token t: id = input_ids[t]; codes[j] = clamp(item_codes[id*8+j], 0, 255);
// out[t, j*16 : j*16+16] = centroids[j, codes[j], :]; zero row if id == 0.
//
// Roofline: 0 FLOPs, ~112 MB of traffic (104.8 MB streaming stores dominate)
// -> ~4.8 us at 23.3 TB/s. Design:
//  * 128 KB centroid table staged into LDS per block via the CDNA5 async
//    global->LDS DMA path (global_load_async_to_lds_b128 / ASYNCcnt).
//  * One token per wave per iteration: lane l -> sub-codebook j=l>>2,
//    float4 quarter q=l&3. Inner gather is ds_load_b128.
//  * Output stores are non-temporal (th:TH_STORE_NT) 512 B/wave
//    global_store_b128 — written once, never re-read; keep L2 free for the
//    random 32 MB item_codes gathers.

#define PQ_M     8
#define PQ_VALS  256
#define PQ_SUB   16                              // floats per sub-embedding
#define CENT_F4  (PQ_M * PQ_VALS * PQ_SUB / 4)   // 8192 float4 = 128 KB
#define BLOCK_T  256
#define GRID_B   512

typedef float f4v __attribute__((ext_vector_type(4)));

__global__ __launch_bounds__(BLOCK_T) void
dpq_gather_kernel(const int* __restrict__ input_ids,
                  const int* __restrict__ item_codes,
                  const float* __restrict__ centroids,
                  float* __restrict__ out,
                  int n_tokens)
{
    extern __shared__ float4 sC[];        // 8192 float4 = 128 KB (dynamic LDS)

    const int tid  = threadIdx.x;
    const int lane = tid & 31;
    const int wave = tid >> 5;

    // ---- Stage centroids (8*256*16 fp32 = 128 KB) into LDS via async DMA ----
    // 16 B per lane per issue; 32 full-EXEC issues per thread cover the table.
    {
        for (int i = tid; i < CENT_F4; i += BLOCK_T) {
            uint32_t gOff = (uint32_t)i * 16u;  // byte offset into centroids
            // Generic shared pointer: low 32 bits ARE the LDS byte address
            // (ISA 10.2).
            uint32_t ldsAddr = (uint32_t)(uintptr_t)(&sC[i]);
            asm volatile(
                "global_load_async_to_lds_b128 %0, %1, %2 offset:0"
                :
                : "v"(ldsAddr), "v"(gOff), "s"(centroids)
                : "memory");
        }
        // Wait for this wave's async transfers, then sync all 8 waves.
        asm volatile("s_wait_asynccnt 0x0" ::: "memory");
    }
    __syncthreads();

    // ---- Gather loop: one token per wave per iteration ----
    const int waves_per_block = BLOCK_T >> 5;
    const int gwave  = blockIdx.x * waves_per_block + wave;
    const int nwaves = gridDim.x * waves_per_block;

    const int j = lane >> 2;   // sub-codebook index 0..7
    const int q = lane & 3;    // which float4 of the 16-float sub-embedding

    for (int t = gwave; t < n_tokens; t += nwaves) {
        // ids are consumed exactly once -> non-temporal load.
        const int id = __builtin_nontemporal_load(input_ids + t);

        // 8 unique 4 B loads per wave (each shared by 4 lanes); keep RT
        // caching: the 32 MB code table is the one thing L2 should hold.
        int code = item_codes[(size_t)id * PQ_M + j];
        code = code < 0 ? 0 : (code > (PQ_VALS - 1) ? (PQ_VALS - 1) : code);

        // LDS gather: 16 B per lane -> ds_load_b128.
        float4 v = sC[((j << 8) + code) * 4 + q];

        f4v r;
        if (id == 0) { r.x = 0.f; r.y = 0.f; r.z = 0.f; r.w = 0.f; }
        else         { r.x = v.x; r.y = v.y; r.z = v.z; r.w = v.w; }

        // 512 B coalesced streaming store per wave -> global_store_b128 (NT).
        __builtin_nontemporal_store(
            r, (f4v*)(out + (size_t)t * (PQ_M * PQ_SUB) + (lane << 2)));
    }
}

extern "C" void kernel_launch(void* const* d_in, const int* in_sizes, int n_in,
                              void* d_out, int out_size, void* d_ws, size_t ws_size,
                              hipStream_t stream) {
    const int*   input_ids  = (const int*)d_in[0];   // (B, S) int32
    const int*   item_codes = (const int*)d_in[1];   // (N, 8) int32
    const float* centroids  = (const float*)d_in[2]; // (8, 256, 16) fp32
    float*       out        = (float*)d_out;         // (B, S, 128) fp32

    const int n_tokens = in_sizes[0];                // B * S

    // 128 KB LDS/block -> 2 blocks per WGP (320 KB); 8 waves/block.
    int grid = GRID_B;
    int max_useful = (n_tokens + (BLOCK_T >> 5) - 1) / (BLOCK_T >> 5);
    if (grid > max_useful) grid = max_useful;
    if (grid < 1) grid = 1;

    dpq_gather_kernel<<<grid, BLOCK_T, CENT_F4 * sizeof(float4), stream>>>(
        input_ids, item_codes, centroids, out, n_tokens);
}